// RecyclingEmbedder_51694226374803
// MI455X (gfx1250) — compile-verified
//
#include <hip/hip_runtime.h>

typedef __attribute__((ext_vector_type(2))) float v2f;
typedef __attribute__((ext_vector_type(8))) float v8f;

#define NRR 512   // NR
#define CZ  128   // c_z
#define CM  256   // c_m
#define LN_EPS 1e-5f

// ---------------------------------------------------------------------------
// z_out = z + layer_norm(z_prev) + (onehot(d2) @ w^T + b)
// One wave per 16 consecutive (i,j) pairs (same i).  The one-hot einsum is an
// exact f32 GEMM: D(16x16) = A(16x16 onehot) * B(16x16 w^T tile) done as four
// chained V_WMMA_F32_16X16X4_F32 per channel tile (8 tiles for 128 channels).
// ---------------------------------------------------------------------------
__global__ __launch_bounds__(256) void ze_kernel(
    const float* __restrict__ z, const float* __restrict__ z_prev,
    const float* __restrict__ x_prev, const float* __restrict__ w,
    const float* __restrict__ bvec, const float* __restrict__ gam,
    const float* __restrict__ bet, float* __restrict__ z_out)
{
    __shared__ float w_lds[CZ * 16];   // w padded to 16 bins (k=15 column = 0)
    __shared__ float x_lds[NRR * 3];

    const int tid = threadIdx.x;
    for (int idx = tid; idx < CZ * 16; idx += 256) {
        int c = idx >> 4, k = idx & 15;
        w_lds[idx] = (k < 15) ? w[c * 15 + k] : 0.0f;
    }
    for (int idx = tid; idx < NRR * 3; idx += 256) x_lds[idx] = x_prev[idx];
    __syncthreads();

    const int lane = tid & 31;
    const int wave = tid >> 5;
    const int g    = blockIdx.x * 8 + wave;   // 0..16383, one wave per group
    const int p0   = g << 4;                  // first (i,j) flat pair index
    const int i    = p0 >> 9;                 // 512 pairs per i => same i
    const int j0   = p0 & (NRR - 1);
    const int half = lane >> 4;               // 0: lanes 0-15, 1: lanes 16-31
    const int l15  = lane & 15;

    // --- distance bin for pair row m = l15 (both halves need the same row) --
    const int j = j0 + l15;
    float dx = x_lds[i * 3 + 0] - x_lds[j * 3 + 0];
    float dy = x_lds[i * 3 + 1] - x_lds[j * 3 + 1];
    float dz = x_lds[i * 3 + 2] - x_lds[j * 3 + 2];
    float d2 = dx * dx + dy * dy + dz * dz;
    int bin = 15;                             // 15 => outside all bins (d = b)
#pragma unroll
    for (int k = 0; k < 15; ++k) {
        float bl = 3.25f + 1.25f * (float)k;
        float lower = bl * bl;
        float bu = 3.25f + 1.25f * (float)(k + 1);
        float upper = (k == 14) ? 1e8f : bu * bu;
        if (d2 > lower && d2 < upper) bin = k;
    }

    // --- A fragments (16x4 f32 each): lanes 0-15 carry K=4q+{0,1},
    //     lanes 16-31 carry K=4q+{2,3}; row M = l15.  k=15 hits a zero column.
    v2f afrag[4];
#pragma unroll
    for (int q = 0; q < 4; ++q) {
        int kb = q * 4 + half * 2;
        afrag[q].x = (bin == kb    ) ? 1.0f : 0.0f;
        afrag[q].y = (bin == kb + 1) ? 1.0f : 0.0f;
    }

    // --- load z_prev rows in D-tile layout; accumulate LN statistics --------
    // element (row p0 + v + 8*half, channel 16*ct + l15)
    const int rowbase = p0 * CZ;
    const int laneoff = half * (8 * CZ) + l15;
    float zp[8][8];                           // [ctile][vgpr]
    float ps[8], ps2[8];
#pragma unroll
    for (int v = 0; v < 8; ++v) { ps[v] = 0.0f; ps2[v] = 0.0f; }
#pragma unroll
    for (int ct = 0; ct < 8; ++ct) {
        int base = rowbase + ct * 16 + laneoff;
#pragma unroll
        for (int v = 0; v < 8; ++v) {
            float t = z_prev[base + v * CZ];
            zp[ct][v] = t;
            ps[v]  += t;
            ps2[v] += t * t;
        }
    }
    // reduce across the 16 lanes of each half (xor 1,2,4,8 never crosses halves)
#pragma unroll
    for (int off = 1; off <= 8; off <<= 1) {
#pragma unroll
        for (int v = 0; v < 8; ++v) {
            ps[v]  += __shfl_xor(ps[v],  off, 32);
            ps2[v] += __shfl_xor(ps2[v], off, 32);
        }
    }
    float mean[8], rstd[8];
#pragma unroll
    for (int v = 0; v < 8; ++v) {
        float mu = ps[v] * (1.0f / CZ);
        float var = ps2[v] * (1.0f / CZ) - mu * mu;
        mean[v] = mu;
        rstd[v] = rsqrtf(var + LN_EPS);
    }

    // --- per channel tile: 4x WMMA for d, then fused add + store ------------
#pragma unroll
    for (int ct = 0; ct < 8; ++ct) {
        const int c = ct * 16 + l15;          // this lane's channel (N = l15)
        v8f acc = {};                         // C = 0
#pragma unroll
        for (int q = 0; q < 4; ++q) {
            int kb = q * 4 + half * 2;        // B: lanes 0-15 K=4q+{0,1}; 16-31 K=4q+{2,3}
            v2f bfrag;
            bfrag.x = w_lds[c * 16 + kb];
            bfrag.y = w_lds[c * 16 + kb + 1];
            acc = __builtin_amdgcn_wmma_f32_16x16x4_f32(
                false, afrag[q], false, bfrag, (short)0, acc, false, false);
        }
        const float gc = gam[c];
        const float bc = bet[c];
        const float bb = bvec[c];
        const int base = rowbase + ct * 16 + laneoff;
#pragma unroll
        for (int v = 0; v < 8; ++v) {         // D row M = v + 8*half
            float zv = z[base + v * CZ];
            float ln = (zp[ct][v] - mean[v]) * rstd[v] * gc + bc;
            z_out[base + v * CZ] = zv + ln + acc[v] + bb;
        }
    }
}

// ---------------------------------------------------------------------------
// m_out[:,0] = m[:,0] + layer_norm(m0_prev); one wave per row (256 channels).
// ---------------------------------------------------------------------------
__global__ __launch_bounds__(256) void m_kernel(
    const float* __restrict__ m, const float* __restrict__ m0_prev,
    const float* __restrict__ gam, const float* __restrict__ bet,
    float* __restrict__ m_out)
{
    const int lane = threadIdx.x & 31;
    const int wave = threadIdx.x >> 5;
    const int r = blockIdx.x * 8 + wave;      // 0..511

    const float4* src = (const float4*)(m0_prev + r * CM);
    float4 a0 = src[lane];
    float4 a1 = src[lane + 32];

    float s  = a0.x + a0.y + a0.z + a0.w + a1.x + a1.y + a1.z + a1.w;
    float ss = a0.x * a0.x + a0.y * a0.y + a0.z * a0.z + a0.w * a0.w
             + a1.x * a1.x + a1.y * a1.y + a1.z * a1.z + a1.w * a1.w;
#pragma unroll
    for (int off = 1; off <= 16; off <<= 1) {
        s  += __shfl_xor(s,  off, 32);
        ss += __shfl_xor(ss, off, 32);
    }
    float mu   = s * (1.0f / CM);
    float var  = ss * (1.0f / CM) - mu * mu;
    float rstd = rsqrtf(var + LN_EPS);

    const float4* mrow = (const float4*)(m + r * CM);   // m[0,0,r,:] = first slab
    const float4* g4 = (const float4*)gam;
    const float4* b4 = (const float4*)bet;
    float4* orow = (float4*)(m_out + r * CM);

    float4 mv0 = mrow[lane],     mv1 = mrow[lane + 32];
    float4 g0  = g4[lane],       g1  = g4[lane + 32];
    float4 b0  = b4[lane],       b1  = b4[lane + 32];

    float4 o0, o1;
    o0.x = mv0.x + (a0.x - mu) * rstd * g0.x + b0.x;
    o0.y = mv0.y + (a0.y - mu) * rstd * g0.y + b0.y;
    o0.z = mv0.z + (a0.z - mu) * rstd * g0.z + b0.z;
    o0.w = mv0.w + (a0.w - mu) * rstd * g0.w + b0.w;
    o1.x = mv1.x + (a1.x - mu) * rstd * g1.x + b1.x;
    o1.y = mv1.y + (a1.y - mu) * rstd * g1.y + b1.y;
    o1.z = mv1.z + (a1.z - mu) * rstd * g1.z + b1.z;
    o1.w = mv1.w + (a1.w - mu) * rstd * g1.w + b1.w;
    orow[lane]      = o0;
    orow[lane + 32] = o1;
}

extern "C" void kernel_launch(void* const* d_in, const int* in_sizes, int n_in,
                              void* d_out, int out_size, void* d_ws, size_t ws_size,
                              hipStream_t stream) {
    (void)in_sizes; (void)n_in; (void)out_size; (void)d_ws; (void)ws_size;
    const float* m       = (const float*)d_in[0];   // [1,128,512,256]
    const float* z       = (const float*)d_in[1];   // [1,512,512,128]
    const float* m0_prev = (const float*)d_in[2];   // [1,512,256]
    const float* z_prev  = (const float*)d_in[3];   // [1,512,512,128]
    const float* x_prev  = (const float*)d_in[4];   // [1,512,3]
    const float* w       = (const float*)d_in[5];   // [128,15]
    const float* b       = (const float*)d_in[6];   // [128]
    const float* gamma_m = (const float*)d_in[7];   // [256]
    const float* beta_m  = (const float*)d_in[8];   // [256]
    const float* gamma_z = (const float*)d_in[9];   // [128]
    const float* beta_z  = (const float*)d_in[10];  // [128]

    float* m_out = (float*)d_out;                       // 128*512*256 = 16777216
    float* z_out = (float*)d_out + 128 * 512 * 256;     // 512*512*128 = 33554432

    // m_out slice nc==0 (first 512*256 floats): LN-add kernel.
    m_kernel<<<64, 256, 0, stream>>>(m, m0_prev, gamma_m, beta_m, m_out);

    // m_out nc>0: straight device-to-device copy (async, capture-safe).
    const size_t skip = (size_t)NRR * CM;               // 131072 floats
    const size_t rest = (size_t)128 * NRR * CM - skip;
    hipMemcpyAsync(m_out + skip, m + skip, rest * sizeof(float),
                   hipMemcpyDeviceToDevice, stream);

    // z path: 262144 pair-rows / 16 per wave / 8 waves per block = 2048 blocks.
    ze_kernel<<<2048, 256, 0, stream>>>(z, z_prev, x_prev, w, b,
                                        gamma_z, beta_z, z_out);
}